// GCNEncoder_3882650437056
// MI455X (gfx1250) — compile-verified
//
#include <hip/hip_runtime.h>

typedef float v2f __attribute__((ext_vector_type(2)));
typedef float v8f __attribute__((ext_vector_type(8)));

// ---------------------------------------------------------------------------
// f32 WMMA 16x16 tile: C(16x16) += A(16x4) * B(4x16) over K, using
// V_WMMA_F32_16X16X4_F32.  One wave per tile.  EXEC is all-ones (tile index
// is wave-uniform).  Layouts per CDNA5 ISA 7.12.2:
//   A: lane 0-15 -> M=lane, {v0=K0,v1=K1}; lane16-31 -> M=lane-16, {K2,K3}
//   B: v0 holds K=0 (lanes0-15,N=lane) / K=2 (lanes16-31); v1 holds K=1/K=3
//   C: VGPR r -> M=r (lanes0-15) and M=r+8 (lanes16-31), N=lane%16
// ---------------------------------------------------------------------------
template <bool TA>
__device__ __forceinline__ void wmma_tile16x16(const float* __restrict__ A, int lda,
                                               const float* __restrict__ B, int ldb,
                                               int K, float* __restrict__ C, int ldc,
                                               const float* __restrict__ bias) {
  const int lane = threadIdx.x & 31;
  const int half = lane >> 4;   // 0 or 1
  const int l16  = lane & 15;
  v8f acc = {};
  for (int k0 = 0; k0 < K; k0 += 4) {
    v2f a, b;
    if (!TA) {  // A row-major, A[m,k] = A[m*lda + k]
      const float* ap = A + (size_t)l16 * lda + (k0 + 2 * half);
      a.x = ap[0];
      a.y = ap[1];
    } else {    // A[m,k] = A[k*lda + m]  (i.e. pass W^T via strides)
      const float* ap = A + (size_t)(k0 + 2 * half) * lda + l16;
      a.x = ap[0];
      a.y = ap[lda];
    }
    const float* bp = B + (size_t)(k0 + 2 * half) * ldb + l16;
    b.x = bp[0];
    b.y = bp[ldb];
    acc = __builtin_amdgcn_wmma_f32_16x16x4_f32(false, a, false, b, (short)0, acc,
                                                false, false);
  }
#pragma unroll
  for (int r = 0; r < 8; ++r) {
    int m = r + 8 * half;
    float v = acc[r];
    if (bias) v += bias[m];
    C[(size_t)m * ldc + l16] = v;
  }
}

// C(MxN) = A(MxK) * B(KxN), all row-major. 4 waves/block, 1 tile/wave.
__global__ void k_gemm_rm(const float* __restrict__ A, const float* __restrict__ B,
                          float* __restrict__ C, int M, int N, int K) {
  int tile   = blockIdx.x * (blockDim.x >> 5) + (threadIdx.x >> 5);
  int tilesN = N >> 4;
  int nt     = (M >> 4) * tilesN;
  if (tile >= nt) return;
  int tm = (tile / tilesN) << 4;
  int tn = (tile % tilesN) << 4;
  wmma_tile16x16<false>(A + (size_t)tm * K, K, B + tn, N, K,
                        C + (size_t)tm * N + tn, N, nullptr);
}

// z_seq[b] (128x128) = W_proj^T (128x128) @ hbar[b] (128x128) + b_proj
__global__ void k_gemm_time(const float* __restrict__ Wproj,
                            const float* __restrict__ hbar,
                            const float* __restrict__ bproj, float* __restrict__ out) {
  int tile  = blockIdx.x * (blockDim.x >> 5) + (threadIdx.x >> 5);
  int batch = tile >> 6;
  int tt    = tile & 63;
  int tm    = (tt >> 3) << 4;
  int tn    = (tt & 7) << 4;
  const float* Bp = hbar + (size_t)batch * 128 * 128 + tn;
  float* Cp       = out + (size_t)batch * 128 * 128 + (size_t)tm * 128 + tn;
  wmma_tile16x16<true>(Wproj + tm, 128, Bp, 128, 128, Cp, 128, bproj + tm);
}

// ---------------------------------------------------------------------------
// x0[n,f] = mean_t h[n,f,t]   (Tp=128; one wave per (n,f) row, float4 loads)
// ---------------------------------------------------------------------------
__global__ void k_mean(const float* __restrict__ h, float* __restrict__ x0, int NF) {
  int warp = (blockIdx.x * blockDim.x + threadIdx.x) >> 5;
  int lane = threadIdx.x & 31;
  if (warp >= NF) return;
  const float4* p = reinterpret_cast<const float4*>(h + (size_t)warp * 128);
  float4 v = p[lane];
  float s  = v.x + v.y + v.z + v.w;
  for (int off = 16; off; off >>= 1) s += __shfl_xor(s, off, 32);
  if (lane == 0) x0[warp] = s * (1.0f / 128.0f);
}

// ---------------------------------------------------------------------------
// Base graph (identical across batches): degrees (weighted + unit, with self
// loop), normalized edge coefs, CSC by target column.  Single block.
// ---------------------------------------------------------------------------
__global__ void k_graph(const int* __restrict__ ei, const float* __restrict__ ew,
                        float* __restrict__ dinvw, float* __restrict__ dinv1,
                        float* __restrict__ normw, float* __restrict__ norm1,
                        int* __restrict__ colstart, int* __restrict__ collist) {
  const int E = 1024, C = 64;
  __shared__ float degw[C];
  __shared__ int   cnt[C], base[C + 1], cur[C];
  __shared__ float sdw[C], sd1[C];
  const int* row = ei;
  const int* col = ei + E;
  int t = threadIdx.x;
  if (t < C) { degw[t] = 1.0f; cnt[t] = 0; }   // self-loop weight 1
  __syncthreads();
  for (int e = t; e < E; e += blockDim.x) {
    atomicAdd(&degw[col[e]], ew[e]);
    atomicAdd(&cnt[col[e]], 1);
  }
  __syncthreads();
  if (t == 0) {
    int s = 0;
    for (int c = 0; c < C; ++c) { base[c] = s; s += cnt[c]; }
    base[C] = s;
  }
  __syncthreads();
  if (t < C) {
    sdw[t] = rsqrtf(degw[t]);
    sd1[t] = rsqrtf((float)(cnt[t] + 1));
    dinvw[t] = sdw[t];
    dinv1[t] = sd1[t];
    cur[t] = base[t];
    colstart[t] = base[t];
    if (t == 0) colstart[C] = base[C];
  }
  __syncthreads();
  for (int e = t; e < E; e += blockDim.x) {
    int c = col[e], r = row[e];
    normw[e] = sdw[r] * ew[e] * sdw[c];
    norm1[e] = sd1[r] * sd1[c];
    int pos = atomicAdd(&cur[c], 1);
    collist[pos] = e;
  }
}

// GCN1 aggregation: x1[n] = selfnorm*xw[n] + sum_in norm*xw[src] + b1
__global__ void k_agg1(const float* __restrict__ xw, const int* __restrict__ ei,
                       const float* __restrict__ normw, const int* __restrict__ colstart,
                       const int* __restrict__ collist, const float* __restrict__ dinvw,
                       const float* __restrict__ b1, float* __restrict__ x1) {
  int n = blockIdx.x, b = n >> 6, c = n & 63, f = threadIdx.x;
  const int* row = ei;
  float dv  = dinvw[c];
  float acc = dv * dv * xw[(size_t)n * 128 + f];
  int e1 = colstart[c + 1];
  for (int p = colstart[c]; p < e1; ++p) {
    int e = collist[p];
    acc += normw[e] * xw[(size_t)(b * 64 + row[e]) * 128 + f];
  }
  x1[(size_t)n * 128 + f] = acc + b1[f];
}

__global__ void k_zero(float* __restrict__ p, int n) {
  int i = blockIdx.x * blockDim.x + threadIdx.x;
  if (i < n) p[i] = 0.0f;
}

// per-feature sums for BN (rows chunked, float atomics into zeroed ws)
__global__ void k_bnstats(const float* __restrict__ x, int rows, float* __restrict__ sum,
                          float* __restrict__ sumsq) {
  int f = threadIdx.x;
  int r0 = blockIdx.x * 64, r1 = r0 + 64;
  if (r1 > rows) r1 = rows;
  float s = 0.f, s2 = 0.f;
  for (int r = r0; r < r1; ++r) {
    float v = x[(size_t)r * 128 + f];
    s += v;
    s2 += v * v;
  }
  atomicAdd(&sum[f], s);
  atomicAdd(&sumsq[f], s2);
}

// BN + PReLU, store y, and per-row dot with Wp (score GEMV fused)
__global__ void k_bn1(const float* __restrict__ x1, const float* __restrict__ sum,
                      const float* __restrict__ sumsq, float invN,
                      const float* __restrict__ g, const float* __restrict__ be,
                      const float* __restrict__ a, const float* __restrict__ Wp,
                      float* __restrict__ y, float* __restrict__ snode) {
  __shared__ float red[128];
  int n = blockIdx.x, f = threadIdx.x;
  float mean = sum[f] * invN;
  float var  = sumsq[f] * invN - mean * mean;
  float v = (x1[(size_t)n * 128 + f] - mean) * rsqrtf(var + 1e-5f) * g[f] + be[f];
  float al = a[0];
  v = v >= 0.f ? v : al * v;
  y[(size_t)n * 128 + f] = v;
  red[f] = v * Wp[f];
  __syncthreads();
  for (int off = 64; off; off >>= 1) {
    if (f < off) red[f] += red[f + off];
    __syncthreads();
  }
  if (f == 0) snode[n] = red[0];
}

// score[n] = tanh(agg(norm1, snode) + bp)
__global__ void k_score(const float* __restrict__ snode, const int* __restrict__ ei,
                        const float* __restrict__ norm1, const int* __restrict__ colstart,
                        const int* __restrict__ collist, const float* __restrict__ dinv1,
                        const float* __restrict__ bp, float* __restrict__ score) {
  int n = blockIdx.x * blockDim.x + threadIdx.x;
  if (n >= 8192) return;
  int b = n >> 6, c = n & 63;
  const int* row = ei;
  float dv  = dinv1[c];
  float acc = dv * dv * snode[n];
  int e1 = colstart[c + 1];
  for (int p = colstart[c]; p < e1; ++p) {
    int e = collist[p];
    acc += norm1[e] * snode[b * 64 + row[e]];
  }
  score[n] = tanhf(acc + bp[0]);
}

// per-batch top-39 of 64 (jax.lax.top_k order: descending, first max on ties),
// softmax weights, perm, and keep-map.  One thread per batch.
__global__ void k_topk(const float* __restrict__ score, float* __restrict__ vals,
                       float* __restrict__ wsoft, int* __restrict__ perm,
                       int* __restrict__ mmap) {
  int b = blockIdx.x * blockDim.x + threadIdx.x;
  if (b >= 128) return;
  const int K = 39;
  const float* sc = score + b * 64;
  unsigned long long used = 0ull;
  float v[K];
  for (int i = 0; i < K; ++i) {
    float bestv = -1e30f;
    int best = 0;
    for (int c = 0; c < 64; ++c) {
      if ((used >> c) & 1ull) continue;
      float s = sc[c];
      if (s > bestv) { bestv = s; best = c; }
    }
    used |= 1ull << best;
    v[i] = bestv;
    vals[b * K + i] = bestv;
    perm[b * K + i] = b * 64 + best;
  }
  float mx = v[0], ssum = 0.f, e[K];
  for (int i = 0; i < K; ++i) { e[i] = expf(v[i] - mx); ssum += e[i]; }
  for (int c = 0; c < 64; ++c) mmap[b * 64 + c] = -1;
  for (int i = 0; i < K; ++i) {
    wsoft[b * K + i] = e[i] / ssum;
    mmap[perm[b * K + i]] = b * K + i;
  }
}

// degree of filtered graph per kept node (self loop + kept in-edges)
__global__ void k_deg2(const int* __restrict__ perm, const int* __restrict__ mmap,
                       const int* __restrict__ ei, const float* __restrict__ ew,
                       const int* __restrict__ colstart, const int* __restrict__ collist,
                       float* __restrict__ dinv2) {
  int i = blockIdx.x * blockDim.x + threadIdx.x;
  if (i >= 4992) return;
  int b = i / 39, c = perm[i] - b * 64;
  const int* row = ei;
  float d = 1.0f;
  int e1 = colstart[c + 1];
  for (int p = colstart[c]; p < e1; ++p) {
    int e = collist[p];
    if (mmap[b * 64 + row[e]] >= 0) d += ew[e];
  }
  dinv2[i] = rsqrtf(d);
}

// xpool[i] = y[perm[i]] * vals[i]
__global__ void k_gate(const float* __restrict__ y, const int* __restrict__ perm,
                       const float* __restrict__ vals, float* __restrict__ xpool) {
  int i = blockIdx.x, f = threadIdx.x;
  xpool[(size_t)i * 128 + f] = y[(size_t)perm[i] * 128 + f] * vals[i];
}

// GCN2 aggregation on the filtered graph
__global__ void k_agg2(const float* __restrict__ xw2, const int* __restrict__ perm,
                       const int* __restrict__ mmap, const int* __restrict__ ei,
                       const float* __restrict__ ew, const int* __restrict__ colstart,
                       const int* __restrict__ collist, const float* __restrict__ dinv2,
                       const float* __restrict__ b2, float* __restrict__ x2) {
  int i = blockIdx.x, f = threadIdx.x;
  int b = i / 39, c = perm[i] - b * 64;
  const int* row = ei;
  float dd  = dinv2[i];
  float acc = dd * dd * xw2[(size_t)i * 128 + f];
  int e1 = colstart[c + 1];
  for (int p = colstart[c]; p < e1; ++p) {
    int e = collist[p];
    int sn = mmap[b * 64 + row[e]];
    if (sn >= 0) acc += dinv2[sn] * ew[e] * dd * xw2[(size_t)sn * 128 + f];
  }
  x2[(size_t)i * 128 + f] = acc + b2[f];
}

// final BN + PReLU -> d_out (first output)
__global__ void k_bn2(const float* __restrict__ x2, const float* __restrict__ sum,
                      const float* __restrict__ sumsq, float invN,
                      const float* __restrict__ g, const float* __restrict__ be,
                      const float* __restrict__ a, float* __restrict__ out) {
  int i = blockIdx.x, f = threadIdx.x;
  float mean = sum[f] * invN;
  float var  = sumsq[f] * invN - mean * mean;
  float v = (x2[(size_t)i * 128 + f] - mean) * rsqrtf(var + 1e-5f) * g[f] + be[f];
  float al = a[0];
  out[(size_t)i * 128 + f] = v >= 0.f ? v : al * v;
}

// hbar[b,f,t] = sum_i w[b,i] * h[b, c_i, f, t]   (coalesced over t)
__global__ void k_hbar(const float* __restrict__ h, const int* __restrict__ perm,
                       const float* __restrict__ wsoft, float* __restrict__ hbar) {
  int b = blockIdx.x >> 7, f = blockIdx.x & 127, t = threadIdx.x;
  float acc = 0.f;
  for (int i = 0; i < 39; ++i) {
    int g = perm[b * 39 + i];  // global node index b*64 + c
    acc += wsoft[b * 39 + i] * h[((size_t)g * 128 + f) * 128 + t];
  }
  hbar[((size_t)b * 128 + f) * 128 + t] = acc;
}

// ---------------------------------------------------------------------------
extern "C" void kernel_launch(void* const* d_in, const int* in_sizes, int n_in,
                              void* d_out, int out_size, void* d_ws, size_t ws_size,
                              hipStream_t stream) {
  const float* h     = (const float*)d_in[0];
  const int*   ei    = (const int*)d_in[1];
  const float* ew    = (const float*)d_in[2];
  const float* Wproj = (const float*)d_in[3];
  const float* bproj = (const float*)d_in[4];
  const float* W1    = (const float*)d_in[5];
  const float* b1    = (const float*)d_in[6];
  const float* g1    = (const float*)d_in[7];
  const float* be1   = (const float*)d_in[8];
  const float* a1    = (const float*)d_in[9];
  const float* Wp    = (const float*)d_in[10];
  const float* bp    = (const float*)d_in[11];
  const float* W2    = (const float*)d_in[12];
  const float* b2    = (const float*)d_in[13];
  const float* g2    = (const float*)d_in[14];
  const float* be2   = (const float*)d_in[15];
  const float* a2    = (const float*)d_in[16];
  float* out = (float*)d_out;

  const int N = 8192, F = 128, E = 1024, C = 64, Bn = 128, Kk = 39, N2 = 4992;

  // workspace carve-up
  char* ws = (char*)d_ws;
  size_t off = 0;
  auto fa = [&](size_t n) { float* p = (float*)(ws + off); off += n * 4; return p; };
  auto ia = [&](size_t n) { int* p = (int*)(ws + off); off += n * 4; return p; };
  float* x0    = fa((size_t)N * F);   // reused as x1 (agg1 output)
  float* xw1   = fa((size_t)N * F);
  float* y     = fa((size_t)N * F);
  float* snode = fa(N);
  float* score = fa(N);
  float* dinvw = fa(C);
  float* dinv1 = fa(C);
  float* normw = fa(E);
  float* norm1 = fa(E);
  float* dinv2 = fa(N2);
  float* vals  = fa((size_t)Bn * Kk);
  float* wsoft = fa((size_t)Bn * Kk);
  float* bns   = fa(512);             // sum1,sq1,sum2,sq2 (128 each)
  float* xpool = fa((size_t)N2 * F);  // reused as x2agg after GEMM2
  float* xw2   = fa((size_t)N2 * F);
  float* hbar  = fa((size_t)Bn * F * 128);
  int* colstart = ia(C + 1);
  int* collist  = ia(E);
  int* perm     = ia(N2);
  int* mmap     = ia(N);
  (void)in_sizes; (void)n_in; (void)out_size; (void)ws_size;

  float* out_x = out;
  float* out_z = out + (size_t)N2 * F;

  k_zero<<<1, 512, 0, stream>>>(bns, 512);
  k_graph<<<1, 256, 0, stream>>>(ei, ew, dinvw, dinv1, normw, norm1, colstart, collist);
  k_mean<<<(N * F) / 8, 256, 0, stream>>>(h, x0, N * F);
  // GCN1 transform: xw1 = x0 @ W1     (8192x128x128, f32 WMMA)
  k_gemm_rm<<<(N / 16) * (F / 16) / 4, 128, 0, stream>>>(x0, W1, xw1, N, F, F);
  k_agg1<<<N, 128, 0, stream>>>(xw1, ei, normw, colstart, collist, dinvw, b1, x0);
  k_bnstats<<<N / 64, 128, 0, stream>>>(x0, N, bns, bns + 128);
  k_bn1<<<N, 128, 0, stream>>>(x0, bns, bns + 128, 1.0f / N, g1, be1, a1, Wp, y, snode);
  k_score<<<N / 256, 256, 0, stream>>>(snode, ei, norm1, colstart, collist, dinv1, bp, score);
  k_topk<<<4, 32, 0, stream>>>(score, vals, wsoft, perm, mmap);
  k_deg2<<<(N2 + 255) / 256, 256, 0, stream>>>(perm, mmap, ei, ew, colstart, collist, dinv2);
  k_gate<<<N2, 128, 0, stream>>>(y, perm, vals, xpool);
  // GCN2 transform: xw2 = xpool @ W2  (4992x128x128, f32 WMMA)
  k_gemm_rm<<<(N2 / 16) * (F / 16) / 4, 128, 0, stream>>>(xpool, W2, xw2, N2, F, F);
  k_agg2<<<N2, 128, 0, stream>>>(xw2, perm, mmap, ei, ew, colstart, collist, dinv2, b2, xpool);
  k_bnstats<<<(N2 + 63) / 64, 128, 0, stream>>>(xpool, N2, bns + 256, bns + 384);
  k_bn2<<<N2, 128, 0, stream>>>(xpool, bns + 256, bns + 384, 1.0f / N2, g2, be2, a2, out_x);
  // time branch: hbar = softmax-weighted sum of kept h rows, then
  // z_seq[b] = W_proj^T @ hbar[b] + b_proj  (f32 WMMA, output layout direct)
  k_hbar<<<Bn * F, 128, 0, stream>>>(h, perm, wsoft, hbar);
  k_gemm_time<<<Bn * 64 / 4, 128, 0, stream>>>(Wproj, hbar, bproj, out_z);
}